// OSRA_71708773974322
// MI455X (gfx1250) — compile-verified
//
#include <hip/hip_runtime.h>
#include <math.h>

typedef __attribute__((ext_vector_type(2))) float v2f;
typedef __attribute__((ext_vector_type(8))) float v8f;

#define NB   32
#define NC   256
#define NHW  4096
#define NK   5
#define ENH_SIZE (NB * NC * NHW)

__device__ __forceinline__ v8f wmma_f32(v2f a, v2f b, v8f c) {
  // D = A(16x4 f32) * B(4x16 f32) + C(16x16 f32), wave32
  return __builtin_amdgcn_wmma_f32_16x16x4_f32(
      /*neg_a=*/false, a, /*neg_b=*/false, b,
      /*c_mod=*/(short)0, c, /*reuse_a=*/false, /*reuse_b=*/false);
}

// ---------------- k0: l2-normalize part queries [K,C] ----------------
__global__ void k0_qnorm(const float* __restrict__ pq, float* __restrict__ qn) {
  __shared__ float red[256];
  int t = threadIdx.x;
  for (int k = 0; k < NK; ++k) {
    float v = pq[k * NC + t];
    red[t] = v * v;
    __syncthreads();
    for (int s = 128; s > 0; s >>= 1) {
      if (t < s) red[t] += red[t + s];
      __syncthreads();
    }
    float nrm = fmaxf(sqrtf(red[0]), 1e-12f);
    __syncthreads();
    qn[k * NC + t] = v / nrm;
  }
}

// ---------------- k1: sim[b,k,n] = qn[k,:]·feat[b,:,n] / (||feat||*temp), masked
__global__ void k1_sim(const float* __restrict__ feat, const float* __restrict__ mask,
                       const float* __restrict__ qn, const float* __restrict__ temp_p,
                       float* __restrict__ sim) {
  __shared__ float sq[NK * NC];
  int t = threadIdx.x;
  int b = blockIdx.y;
  int n = blockIdx.x * 256 + t;
  for (int i = t; i < NK * NC; i += 256) sq[i] = qn[i];
  __syncthreads();
  const float* fb = feat + (size_t)b * NC * NHW + n;
  float ss = 0.f;
  float a[NK] = {0.f, 0.f, 0.f, 0.f, 0.f};
#pragma unroll 4
  for (int c = 0; c < NC; ++c) {
    float f = fb[(size_t)c * NHW];
    ss += f * f;
#pragma unroll
    for (int k = 0; k < NK; ++k) a[k] += f * sq[k * NC + c];
  }
  float temp = fmaxf(temp_p[0], 0.01f);
  float inv = 1.f / (fmaxf(sqrtf(ss), 1e-12f) * temp);
  bool valid = mask[(size_t)b * NHW + n] > 0.f;
  float neg = -__builtin_inff();
#pragma unroll
  for (int k = 0; k < NK; ++k)
    sim[((size_t)b * NK + k) * NHW + n] = valid ? a[k] * inv : neg;
}

// ---------------- k2: masked softmax over HW, written in place (attn) --------
__global__ void k2_softmax(float* __restrict__ sim) {
  __shared__ float red[256];
  int t = threadIdx.x;
  float* s = sim + (size_t)blockIdx.x * NHW;
  float m = -__builtin_inff();
  for (int i = t; i < NHW; i += 256) m = fmaxf(m, s[i]);
  red[t] = m;
  __syncthreads();
  for (int st = 128; st > 0; st >>= 1) {
    if (t < st) red[t] = fmaxf(red[t], red[t + st]);
    __syncthreads();
  }
  float mv = red[0];
  __syncthreads();
  bool any = mv > -__builtin_inff();
  float mu = any ? mv : 0.f;
  float p = 0.f;
  for (int i = t; i < NHW; i += 256) p += expf(s[i] - mu);
  red[t] = p;
  __syncthreads();
  for (int st = 128; st > 0; st >>= 1) {
    if (t < st) red[t] += red[t + st];
    __syncthreads();
  }
  float inv = any ? (1.f / red[0]) : 0.f;
  __syncthreads();
  for (int i = t; i < NHW; i += 256) s[i] = expf(s[i] - mu) * inv;
}

// ---------------- k3: part_tokens[b,k,c] = Σ_n attn[b,k,n]·feat[b,c,n] -------
__global__ void k3_tokens(const float* __restrict__ feat, const float* __restrict__ attn,
                          float* __restrict__ pt) {
  int lane = threadIdx.x & 31;
  int wave = threadIdx.x >> 5;           // 8 waves: one channel row each
  int c = blockIdx.x * 8 + wave;
  int b = blockIdx.y;
  const float* fb = feat + ((size_t)b * NC + c) * NHW;
  const float* at = attn + (size_t)b * NK * NHW;
  float acc[NK] = {0.f, 0.f, 0.f, 0.f, 0.f};
  for (int n = lane; n < NHW; n += 32) {
    float f = fb[n];
#pragma unroll
    for (int k = 0; k < NK; ++k) acc[k] += f * at[k * NHW + n];
  }
#pragma unroll
  for (int k = 0; k < NK; ++k) {
    float v = acc[k];
    for (int off = 16; off > 0; off >>= 1) v += __shfl_down(v, off);
    if (lane == 0) pt[((size_t)b * NK + k) * NC + c] = v;
  }
}

// ---------------- k4: diversity loss (single block -> deterministic) ---------
__global__ void k4_div(const float* __restrict__ pt, const float* __restrict__ margin,
                       float* __restrict__ loss_out) {
  __shared__ float nt[NK][NC];
  __shared__ float red[256];
  int t = threadIdx.x;
  float m = fminf(fmaxf(margin[0], 0.f), 0.5f);
  const int pk[10] = {0, 0, 0, 0, 1, 1, 1, 2, 2, 3};
  const int pj[10] = {1, 2, 3, 4, 2, 3, 4, 3, 4, 4};
  float acc = 0.f;
  for (int b = 0; b < NB; ++b) {
    for (int k = 0; k < NK; ++k) {
      float v = pt[((size_t)b * NK + k) * NC + t];
      red[t] = v * v;
      __syncthreads();
      for (int s = 128; s > 0; s >>= 1) {
        if (t < s) red[t] += red[t + s];
        __syncthreads();
      }
      float nrm = fmaxf(sqrtf(red[0]), 1e-12f);
      __syncthreads();
      nt[k][t] = v / nrm;
    }
    __syncthreads();
    for (int p = 0; p < 10; ++p) {
      red[t] = nt[pk[p]][t] * nt[pj[p]][t];
      __syncthreads();
      for (int s = 128; s > 0; s >>= 1) {
        if (t < s) red[t] += red[t + s];
        __syncthreads();
      }
      if (t == 0) acc += 2.f * fmaxf(red[0] - m, 0.f);
      __syncthreads();
    }
  }
  if (t == 0) *loss_out = acc / 640.00001f;  // B*K*(K-1) + 1e-5
}

// ---------------- k5: per-batch WMMA GEMMs -> QK, OVt, ck --------------------
// All A matrices are staged zero-padded (16 rows) in LDS so the WMMA inner
// loops contain NO lane-predicated loads (ds_load_b64 + global_load_b64 + wmma
// only; EXEC stays all-1s by construction).
// Kt = PT @ k_w^T + k_b ; Vt = PT @ v_w^T + v_b (pad rows stored as 0)
// QK = ((scale⊙Kt) @ q_w) / 16 ; OVt = Vt @ out_w^T ; ck = (shift·Kt) / 16
__global__ void k5_gemms(const float* __restrict__ pt,
                         const float* __restrict__ k_w, const float* __restrict__ k_b,
                         const float* __restrict__ v_w, const float* __restrict__ v_b,
                         const float* __restrict__ q_w, const float* __restrict__ out_w,
                         const float* __restrict__ bn_w, const float* __restrict__ bn_b,
                         const float* __restrict__ bn_mean, const float* __restrict__ bn_var,
                         float* __restrict__ qk, float* __restrict__ ovt,
                         float* __restrict__ ck) {
  __shared__ float PtP[16 * NC];   // zero-padded part tokens
  __shared__ float KtP[16 * NC];   // pad rows written as 0
  __shared__ float VtP[16 * NC];   // pad rows written as 0
  __shared__ float sScale[NC];
  __shared__ float sShift[NC];
  int b = blockIdx.x;
  int t = threadIdx.x;
  int wave = t >> 5;
  int lane = t & 31;
  int m = lane & 15;
  int col = lane & 15;
  int kb = (lane >> 4) * 2;   // K-pair offset for both A and B fragments
  int hi = lane >> 4;

  {
    float s = bn_w[t] / sqrtf(bn_var[t] + 1e-5f);
    sScale[t] = s;
    sShift[t] = bn_b[t] - bn_mean[t] * s;
  }
  // cooperative, coalesced, zero-padded stage of PT (rows >= 5 are zero)
  for (int i = t; i < 16 * NC; i += 256)
    PtP[i] = (i < NK * NC) ? pt[(size_t)b * NK * NC + i] : 0.f;
  __syncthreads();

  // ---- pass 1: Kt and Vt tiles (32 wave-jobs, wave-uniform) ----
  for (int it = 0; it < 4; ++it) {
    int job = wave + 8 * it;
    bool isK = job < 16;
    int n0 = (isK ? job : job - 16) * 16;
    const float* Wm = isK ? k_w : v_w;
    const float* bias = isK ? k_b : v_b;
    v8f acc = {0.f, 0.f, 0.f, 0.f, 0.f, 0.f, 0.f, 0.f};
    int o = n0 + col;
#pragma unroll 4
    for (int kk = 0; kk < 64; ++kk) {
      int c4 = kk * 4;
      v2f a, bb;
      a.x = PtP[m * NC + c4 + kb];              // unconditional ds_load
      a.y = PtP[m * NC + c4 + kb + 1];
      bb.x = Wm[(size_t)o * NC + c4 + kb];      // B[c,o] = W[o,c]
      bb.y = Wm[(size_t)o * NC + c4 + kb + 1];
      acc = wmma_f32(a, bb, acc);
    }
    float bv = bias[o];
    float* dst = isK ? KtP : VtP;
#pragma unroll
    for (int r = 0; r < 8; ++r) {
      int row = r + 8 * hi;
      dst[row * NC + o] = (row < NK) ? (acc[r] + bv) : 0.f;  // pad rows -> 0
    }
  }
  __syncthreads();

  // ---- ck[k] = Σ_o shift[o]·Kt[k,o] / 16 ----
  if (t < NK * 32) {
    int k = t >> 5, l = t & 31;
    float p = 0.f;
    for (int o = l; o < NC; o += 32) p += sShift[o] * KtP[k * NC + o];
    for (int off = 16; off > 0; off >>= 1) p += __shfl_down(p, off);
    if (l == 0) ck[b * NK + k] = p * 0.0625f;
  }

  // ---- pass 2: QK and OVt tiles ----
  for (int it = 0; it < 4; ++it) {
    int job = wave + 8 * it;
    bool isQ = job < 16;
    int n0 = (isQ ? job : job - 16) * 16;
    v8f acc = {0.f, 0.f, 0.f, 0.f, 0.f, 0.f, 0.f, 0.f};
#pragma unroll 4
    for (int kk = 0; kk < 64; ++kk) {
      int c4 = kk * 4;
      int i0 = c4 + kb, i1 = c4 + kb + 1;
      v2f a, bb;
      if (isQ) {  // A' = scale⊙Kt (reduce over o), B = q_w[o,c]
        a.x = KtP[m * NC + i0] * sScale[i0];
        a.y = KtP[m * NC + i1] * sScale[i1];
        bb.x = q_w[(size_t)i0 * NC + n0 + col];
        bb.y = q_w[(size_t)i1 * NC + n0 + col];
      } else {    // A = Vt (reduce over c), B[c,o] = out_w[o,c]
        a.x = VtP[m * NC + i0];
        a.y = VtP[m * NC + i1];
        int o = n0 + col;
        bb.x = out_w[(size_t)o * NC + i0];
        bb.y = out_w[(size_t)o * NC + i1];
      }
      acc = wmma_f32(a, bb, acc);
    }
#pragma unroll
    for (int r = 0; r < 8; ++r) {
      int row = r + 8 * hi;
      if (row < NK) {
        if (isQ) qk[((size_t)b * NK + row) * NC + n0 + col] = acc[r] * 0.0625f;
        else     ovt[((size_t)b * NK + row) * NC + n0 + col] = acc[r];
      }
    }
  }
}

// ---------------- k6: logits + softmax(K) + rank-5 output + residual ---------
__global__ void k6_refine(const float* __restrict__ feat, const float* __restrict__ qk,
                          const float* __restrict__ ovt, const float* __restrict__ ck,
                          const float* __restrict__ out_b, const float* __restrict__ gamma,
                          float* __restrict__ outp) {
  __shared__ float sQK[NK * NC];
  __shared__ float sOV[NK * NC];
  __shared__ float sOB[NC];
  __shared__ float sCK[NK];
  int t = threadIdx.x;
  int b = blockIdx.y;
  int n = blockIdx.x * 256 + t;
  for (int i = t; i < NK * NC; i += 256) {
    sQK[i] = qk[(size_t)b * NK * NC + i];
    sOV[i] = ovt[(size_t)b * NK * NC + i];
  }
  sOB[t] = out_b[t];
  if (t < NK) sCK[t] = ck[b * NK + t];
  __syncthreads();

  const float* fb = feat + (size_t)b * NC * NHW + n;
  float l[NK];
#pragma unroll
  for (int k = 0; k < NK; ++k) l[k] = sCK[k];
#pragma unroll 4
  for (int c = 0; c < NC; ++c) {
    float f = fb[(size_t)c * NHW];
#pragma unroll
    for (int k = 0; k < NK; ++k) l[k] += f * sQK[k * NC + c];
  }
  float mx = l[0];
#pragma unroll
  for (int k = 1; k < NK; ++k) mx = fmaxf(mx, l[k]);
  float w[NK], s = 0.f;
#pragma unroll
  for (int k = 0; k < NK; ++k) { w[k] = expf(l[k] - mx); s += w[k]; }
  float inv = 1.f / s;
#pragma unroll
  for (int k = 0; k < NK; ++k) w[k] *= inv;

  float g = gamma[0];
  float* ob = outp + (size_t)b * NC * NHW + n;
#pragma unroll 4
  for (int o = 0; o < NC; ++o) {
    float f = fb[(size_t)o * NHW];
    float a = sOB[o];
#pragma unroll
    for (int k = 0; k < NK; ++k) a += w[k] * sOV[k * NC + o];
    ob[(size_t)o * NHW] = f + g * a;
  }
}

extern "C" void kernel_launch(void* const* d_in, const int* in_sizes, int n_in,
                              void* d_out, int out_size, void* d_ws, size_t ws_size,
                              hipStream_t stream) {
  const float* supp_feat    = (const float*)d_in[0];
  const float* supp_mask    = (const float*)d_in[1];
  const float* part_queries = (const float*)d_in[2];
  const float* temperature  = (const float*)d_in[3];
  const float* margin       = (const float*)d_in[4];
  const float* q_w          = (const float*)d_in[5];
  const float* bn_w         = (const float*)d_in[6];
  const float* bn_b         = (const float*)d_in[7];
  const float* bn_mean      = (const float*)d_in[8];
  const float* bn_var       = (const float*)d_in[9];
  const float* k_w          = (const float*)d_in[10];
  const float* k_b          = (const float*)d_in[11];
  const float* v_w          = (const float*)d_in[12];
  const float* v_b          = (const float*)d_in[13];
  const float* out_w        = (const float*)d_in[14];
  const float* out_b        = (const float*)d_in[15];
  const float* gamma        = (const float*)d_in[16];

  float* out  = (float*)d_out;
  float* enh  = out;                       // [B,C,H,W]
  float* loss = out + ENH_SIZE;            // scalar
  float* pt   = loss + 1;                  // [B,K,C]

  float* ws  = (float*)d_ws;               // ~2.96 MB used
  float* qn  = ws;                         // K*C          = 1280
  float* sim = qn + NK * NC;               // B*K*HW       = 655360 (becomes attn)
  float* qk  = sim + (size_t)NB * NK * NHW;// B*K*C        = 40960
  float* ovt = qk + NB * NK * NC;          // B*K*C        = 40960
  float* ckw = ovt + NB * NK * NC;         // B*K          = 160

  k0_qnorm  <<<1,               256, 0, stream>>>(part_queries, qn);
  k1_sim    <<<dim3(16, NB),    256, 0, stream>>>(supp_feat, supp_mask, qn, temperature, sim);
  k2_softmax<<<NB * NK,         256, 0, stream>>>(sim);
  k3_tokens <<<dim3(NC / 8, NB),256, 0, stream>>>(supp_feat, sim, pt);
  k4_div    <<<1,               256, 0, stream>>>(pt, margin, loss);
  k5_gemms  <<<NB,              256, 0, stream>>>(pt, k_w, k_b, v_w, v_b, q_w, out_w,
                                                  bn_w, bn_b, bn_mean, bn_var, qk, ovt, ckw);
  k6_refine <<<dim3(16, NB),    256, 0, stream>>>(supp_feat, qk, ovt, ckw, out_b, gamma, enh);
}